// Interaction_Net_52690658787334
// MI455X (gfx1250) — compile-verified
//
#include <hip/hip_runtime.h>
#include <math.h>

// ---------------------------------------------------------------------------
// Interaction network block for MI455X (gfx1250, wave32, WMMA f32 path).
// Inputs (setup_inputs dict flattened, params in insertion order):
//  0 x_source[900,256] 1 x_target[100000,256] 2 edge_src[E] 3 edge_dst[E]
//  4 Wq 5 bq 6 Wk 7 bk 8 Wv 9 bv 10 Wih 11 bih 12 Whh 13 bhh 14 Wo 15 bo
// 16 W1[256,1024] 17 b1 18 W2[1024,256] 19 b2 20 n1_g 21 n1_b 22 n3_g 23 n3_b
// ---------------------------------------------------------------------------

typedef float v2f __attribute__((ext_vector_type(2)));
typedef float v8f __attribute__((ext_vector_type(8)));

#define DIM   256
#define NH    8
#define DHEAD 32
#define NSRC  900
#define NTGT  100000
#define NEDGE 300000

// ---------------------------------------------------------------- utilities
__global__ void fill_f32(float* __restrict__ p, float v, long n) {
    long i = (long)blockIdx.x * blockDim.x + threadIdx.x;
    if (i < n) p[i] = v;
}

__device__ inline void atomicMaxFloat(float* addr, float val) {
    if ((__float_as_uint(val) >> 31) == 0)
        atomicMax((int*)addr, __float_as_int(val));
    else
        atomicMin((unsigned int*)addr, __float_as_uint(val));
}

// Async copy of 16 bytes global -> LDS, tracked on ASYNCcnt (CDNA5 path).
__device__ inline void async_g2l_b128(void* lds_ptr, const float* gptr) {
    unsigned lo = (unsigned)(((size_t)lds_ptr) & 0xffffffffu);  // LDS offset
    unsigned long long ga = (unsigned long long)(size_t)gptr;
    asm volatile("global_load_async_to_lds_b128 %0, %1, off"
                 :: "v"(lo), "v"(ga) : "memory");
}
__device__ inline void async_wait0() {
    asm volatile("s_wait_asynccnt 0x0" ::: "memory");
}

// Repack W[K,N] so each B fragment is one aligned b64:
// Wp[((k4*2 + g)*N + col)*2 + j] = W[(4*k4 + 2*g + j)*N + col]
__global__ void pack_w(const float* __restrict__ W, float* __restrict__ Wp,
                       int K, int N) {
    long i = (long)blockIdx.x * blockDim.x + threadIdx.x;  // pair index
    long npairs = (long)K * N / 2;
    if (i >= npairs) return;
    int k4  = (int)(i / (2 * N));
    int rem = (int)(i % (2 * N));
    int g   = rem / N;
    int col = rem % N;
    Wp[i * 2 + 0] = W[(size_t)(4 * k4 + 2 * g + 0) * N + col];
    Wp[i * 2 + 1] = W[(size_t)(4 * k4 + 2 * g + 1) * N + col];
}

// Row LayerNorm over 256 columns, one wave (32 lanes x 8 elems) per row.
__global__ __launch_bounds__(256)
void ln_rows(const float* __restrict__ X, const float* __restrict__ gamma,
             const float* __restrict__ beta, float* __restrict__ Y, int M) {
    int lane = threadIdx.x & 31;
    int wave = threadIdx.x >> 5;
    int row  = blockIdx.x * 8 + wave;
    if (row >= M) return;
    const float* xp = X + (size_t)row * DIM + lane * 8;
    float4 a = ((const float4*)xp)[0];
    float4 b = ((const float4*)xp)[1];
    float s  = a.x + a.y + a.z + a.w + b.x + b.y + b.z + b.w;
    float s2 = a.x*a.x + a.y*a.y + a.z*a.z + a.w*a.w
             + b.x*b.x + b.y*b.y + b.z*b.z + b.w*b.w;
    #pragma unroll
    for (int off = 1; off < 32; off <<= 1) {
        s  += __shfl_xor(s,  off);
        s2 += __shfl_xor(s2, off);
    }
    float mu  = s * (1.0f / DIM);
    float var = s2 * (1.0f / DIM) - mu * mu;
    float inv = rsqrtf(var + 1e-5f);
    int c = lane * 8;
    float* yp = Y + (size_t)row * DIM + c;
    float4 g0 = ((const float4*)(gamma + c))[0], g1 = ((const float4*)(gamma + c))[1];
    float4 e0 = ((const float4*)(beta  + c))[0], e1 = ((const float4*)(beta  + c))[1];
    float4 o0, o1;
    o0.x = (a.x - mu) * inv * g0.x + e0.x; o0.y = (a.y - mu) * inv * g0.y + e0.y;
    o0.z = (a.z - mu) * inv * g0.z + e0.z; o0.w = (a.w - mu) * inv * g0.w + e0.w;
    o1.x = (b.x - mu) * inv * g1.x + e1.x; o1.y = (b.y - mu) * inv * g1.y + e1.y;
    o1.z = (b.z - mu) * inv * g1.z + e1.z; o1.w = (b.w - mu) * inv * g1.w + e1.w;
    ((float4*)yp)[0] = o0;
    ((float4*)yp)[1] = o1;
}

// ------------------------------------------------------------ WMMA f32 GEMM
// C[M,N] = epilogue(A[M,K] @ W[K,N] + bias), K=256, N=256, W pre-packed.
// Block = 256 threads (8 waves), 64x256 tile. A (64 KB) async-staged to LDS,
// XOR-swizzled at float4 granularity for conflict-free fragment reads.
// Each wave owns 64x32 = 4x2 tiles: A frags reused 2x, B frags 4x, B frag =
// one b64 from the packed layout.
// EPI 0: none | 1: +X1 residual | 2: relu | 3: gate u = X2*sigmoid(acc+X1)
template <int EPI>
__global__ __launch_bounds__(256)
void gemm_wmma_f32(const float* __restrict__ A, const float* __restrict__ Wp,
                   const float* __restrict__ bias,
                   const float* __restrict__ X1, const float* __restrict__ X2,
                   float* __restrict__ Cout, int M, int K, int N) {
    __shared__ float4 Alds4[64 * 64];   // 64 rows x 64 float4 = 64 KB
    const int tid  = threadIdx.x;
    const int lane = tid & 31;
    const int wave = tid >> 5;
    const int g = lane >> 4;            // lane half (0/1)
    const int r = lane & 15;
    const int m0 = blockIdx.x * 64;

    // ---- stage A tile: 16 async b128 transfers per thread, swizzled dest
    #pragma unroll
    for (int j = 0; j < 16; ++j) {
        int f4  = tid + j * 256;        // 0..4095
        int row = f4 >> 6;
        int c4  = f4 & 63;
        int grow = m0 + row; if (grow > M - 1) grow = M - 1;   // tail clamp
        async_g2l_b128(&Alds4[row * 64 + (c4 ^ (row & 15))],
                       A + (size_t)grow * K + c4 * 4);
    }
    async_wait0();
    __syncthreads();

    // ---- main K loop
    const int nw = wave * 32;           // wave's N base (N==256, 8 waves)
    v8f acc[4][2] = {};
    for (int k = 0; k < K; k += 4) {
        v2f a[4];
        #pragma unroll
        for (int mt = 0; mt < 4; ++mt) {
            int row  = mt * 16 + r;
            int idx4 = row * 64 + ((k >> 2) ^ (row & 15));
            a[mt] = *(const v2f*)((const char*)Alds4 + (size_t)idx4 * 16 + 8 * g);
        }
        const float* wrow = Wp + ((size_t)(k >> 2) * 2 + g) * (2 * N);
        #pragma unroll
        for (int nt = 0; nt < 2; ++nt) {
            int col = nw + nt * 16 + r;
            v2f b = *(const v2f*)(wrow + (size_t)col * 2);
            #pragma unroll
            for (int mt = 0; mt < 4; ++mt)
                acc[mt][nt] = __builtin_amdgcn_wmma_f32_16x16x4_f32(
                    false, a[mt], false, b, (short)0, acc[mt][nt], false, false);
        }
    }

    // ---- epilogue
    #pragma unroll
    for (int nt = 0; nt < 2; ++nt) {
        const int col = nw + nt * 16 + r;
        const float bv = bias[col];
        #pragma unroll
        for (int mt = 0; mt < 4; ++mt) {
            #pragma unroll
            for (int v = 0; v < 8; ++v) {
                int row = m0 + mt * 16 + v + 8 * g;
                if (row < M) {
                    size_t idx = (size_t)row * N + col;
                    float x = acc[mt][nt][v] + bv;
                    if (EPI == 1) x += X1[idx];
                    if (EPI == 2) x = fmaxf(x, 0.0f);
                    if (EPI == 3) {
                        float gate = 1.0f / (1.0f + __expf(-(x + X1[idx])));
                        x = X2[idx] * gate;
                    }
                    Cout[idx] = x;
                }
            }
        }
    }
}

// ---------------------------------------------------------------- edge work
__global__ void make_mask(const float* __restrict__ Vs, float* __restrict__ maskS) {
    int i = blockIdx.x * blockDim.x + threadIdx.x;   // (src, head)
    if (i >= NSRC * NH) return;
    const float* vp = Vs + (size_t)i * DHEAD;
    float s = 0.0f;
    #pragma unroll
    for (int j = 0; j < DHEAD; ++j) s += vp[j];
    maskS[i] = (s != 0.0f) ? 1.0f : 0.0f;
}

// Pass A: wave per edge -> alpha[E,H], running per-(tgt,head) max.
__global__ __launch_bounds__(256)
void edge_alpha(const float* __restrict__ Qt, const float* __restrict__ Ks,
                const int* __restrict__ esrc, const int* __restrict__ edst,
                float* __restrict__ alpha, float* __restrict__ amax) {
    int lane = threadIdx.x & 31;
    int wave = threadIdx.x >> 5;
    int e = blockIdx.x * 8 + wave;
    if (e >= NEDGE) return;
    int s = esrc[e], t = edst[e];
    const float4* q = (const float4*)(Qt + (size_t)t * DIM + lane * 8);
    const float4* k = (const float4*)(Ks + (size_t)s * DIM + lane * 8);
    float4 q0 = q[0], q1 = q[1], k0 = k[0], k1 = k[1];
    float p = q0.x*k0.x + q0.y*k0.y + q0.z*k0.z + q0.w*k0.w
            + q1.x*k1.x + q1.y*k1.y + q1.z*k1.z + q1.w*k1.w;
    p += __shfl_xor(p, 1);
    p += __shfl_xor(p, 2);               // 4-lane group == one head (32 dims)
    if ((lane & 3) == 0) {
        int h = lane >> 2;
        float av = p * 0.17677669529663687f;   // 1/sqrt(32)
        alpha[(size_t)e * NH + h] = av;
        atomicMaxFloat(&amax[(size_t)t * NH + h], av);
    }
}

// Pass B: exp(alpha - max), accumulate per-(tgt,head) denominator.
__global__ void edge_exp(const int* __restrict__ edst, float* __restrict__ alpha,
                         const float* __restrict__ amax, float* __restrict__ denom) {
    long i = (long)blockIdx.x * blockDim.x + threadIdx.x;
    if (i >= (long)NEDGE * NH) return;
    int e = (int)(i >> 3), h = (int)(i & 7);
    int t = edst[e];
    float ex = __expf(alpha[i] - amax[(size_t)t * NH + h]);
    alpha[i] = ex;
    atomicAdd(&denom[(size_t)t * NH + h], ex);
}

// Pass C: agg[t,:] += Vs[s,:] * softmax_weight * mask  (wave per edge).
__global__ __launch_bounds__(256)
void edge_agg(const int* __restrict__ esrc, const int* __restrict__ edst,
              const float* __restrict__ alpha, const float* __restrict__ denom,
              const float* __restrict__ maskS, const float* __restrict__ Vs,
              float* __restrict__ agg) {
    int lane = threadIdx.x & 31;
    int wave = threadIdx.x >> 5;
    int e = blockIdx.x * 8 + wave;
    if (e >= NEDGE) return;
    int s = esrc[e], t = edst[e];
    int h = lane >> 2;
    float w = alpha[(size_t)e * NH + h] /
              (denom[(size_t)t * NH + h] + 1e-16f) * maskS[s * NH + h];
    const float* vp = Vs + (size_t)s * DIM + lane * 8;
    float* ap = agg + (size_t)t * DIM + lane * 8;
    #pragma unroll
    for (int j = 0; j < 8; ++j) atomicAdd(&ap[j], vp[j] * w);
}

// ------------------------------------------------- fused MLP (W1-relu-W2)
// y = xt + relu(h @ W1 + b1) @ W2 + b2.
// Block = 256 threads (8 waves), 32 rows. Hidden tile 32x1024 f32 lives in
// 128 KB LDS (XOR-swizzled at float4 granularity -> conflict-free fragment
// reads in phase 2). The 32x256 LN-input tile is async-staged into the FRONT
// of the same LDS buffer (fully consumed before the hidden epilogue
// overwrites it; barriers separate the phases). W1/W2 are pre-packed so B
// fragments are single b64 loads.
__global__ __launch_bounds__(256)
void mlp_fused(const float* __restrict__ Hn, const float* __restrict__ W1p,
               const float* __restrict__ b1, const float* __restrict__ W2p,
               const float* __restrict__ b2, float* __restrict__ Y, int M) {
    __shared__ float4 lds4[32 * 256];   // 128 KB: [0,2048) doubles as A stage
    float* ldsf = (float*)lds4;
    const int tid  = threadIdx.x;
    const int lane = tid & 31;
    const int wave = tid >> 5;          // 8 waves
    const int g = lane >> 4, r = lane & 15;
    const int m0 = blockIdx.x * 32;

    // ---- async-stage LN-input tile (32 x 256 = 32 KB) into lds4[0..2047]
    #pragma unroll
    for (int j = 0; j < 8; ++j) {
        int f4  = tid + j * 256;        // 0..2047
        int row = f4 >> 6;
        int c4  = f4 & 63;
        int grow = m0 + row; if (grow > M - 1) grow = M - 1;
        async_g2l_b128(&lds4[row * 64 + (c4 ^ (row & 15))],
                       Hn + (size_t)grow * 256 + c4 * 4);
    }
    async_wait0();
    __syncthreads();

    // ---- phase 1: hidden = relu(A @ W1 + b1); wave covers 128 of 1024 cols
    const int nb1 = wave * 128;
    v8f acc1[2][8] = {};
    for (int k = 0; k < 256; k += 4) {
        v2f a[2];
        #pragma unroll
        for (int mt = 0; mt < 2; ++mt) {
            int row  = mt * 16 + r;
            int idx4 = row * 64 + ((k >> 2) ^ (row & 15));
            a[mt] = *(const v2f*)((const char*)lds4 + (size_t)idx4 * 16 + 8 * g);
        }
        const float* wrow = W1p + ((size_t)(k >> 2) * 2 + g) * 2048;
        #pragma unroll
        for (int nt = 0; nt < 8; ++nt) {
            int col = nb1 + nt * 16 + r;
            v2f b = *(const v2f*)(wrow + (size_t)col * 2);
            #pragma unroll
            for (int mt = 0; mt < 2; ++mt)
                acc1[mt][nt] = __builtin_amdgcn_wmma_f32_16x16x4_f32(
                    false, a[mt], false, b, (short)0, acc1[mt][nt], false, false);
        }
    }
    __syncthreads();   // all waves done reading the A-stage region

    // write hidden tile, swizzled: float 'col' of 'row' at
    // ldsf[(row*256 + ((col>>2) ^ (row&15)))*4 + (col&3)]
    #pragma unroll
    for (int nt = 0; nt < 8; ++nt) {
        int col = nb1 + nt * 16 + r;
        float bv = b1[col];
        #pragma unroll
        for (int mt = 0; mt < 2; ++mt) {
            #pragma unroll
            for (int v = 0; v < 8; ++v) {
                int row = mt * 16 + v + 8 * g;
                int idx4 = row * 256 + ((col >> 2) ^ (row & 15));
                ldsf[idx4 * 4 + (col & 3)] = fmaxf(acc1[mt][nt][v] + bv, 0.0f);
            }
        }
    }
    __syncthreads();

    // ---- phase 2: y = xt + hidden @ W2 + b2; wave covers 32 of 256 cols
    const int nb2 = wave * 32;
    v8f acc2[2][2] = {};
    for (int k = 0; k < 1024; k += 4) {
        v2f a[2];
        #pragma unroll
        for (int mt = 0; mt < 2; ++mt) {
            int row  = mt * 16 + r;
            int idx4 = row * 256 + ((k >> 2) ^ (row & 15));
            a[mt] = *(const v2f*)((const char*)lds4 + (size_t)idx4 * 16 + 8 * g);
        }
        const float* wrow = W2p + ((size_t)(k >> 2) * 2 + g) * 512;
        #pragma unroll
        for (int nt = 0; nt < 2; ++nt) {
            int col = nb2 + nt * 16 + r;
            v2f b = *(const v2f*)(wrow + (size_t)col * 2);
            #pragma unroll
            for (int mt = 0; mt < 2; ++mt)
                acc2[mt][nt] = __builtin_amdgcn_wmma_f32_16x16x4_f32(
                    false, a[mt], false, b, (short)0, acc2[mt][nt], false, false);
        }
    }
    #pragma unroll
    for (int nt = 0; nt < 2; ++nt) {
        int col = nb2 + nt * 16 + r;
        float bv = b2[col];
        #pragma unroll
        for (int mt = 0; mt < 2; ++mt) {
            #pragma unroll
            for (int v = 0; v < 8; ++v) {
                int row = m0 + mt * 16 + v + 8 * g;
                if (row < M) {
                    size_t idx = (size_t)row * 256 + col;
                    Y[idx] = Y[idx] + acc2[mt][nt][v] + bv;
                }
            }
        }
    }
}

// ------------------------------------------------------------------ launcher
extern "C" void kernel_launch(void* const* d_in, const int* in_sizes, int n_in,
                              void* d_out, int out_size, void* d_ws, size_t ws_size,
                              hipStream_t stream) {
    const float* x_src = (const float*)d_in[0];
    const float* x_tgt = (const float*)d_in[1];
    const int*   esrc  = (const int*)d_in[2];
    const int*   edst  = (const int*)d_in[3];
    const float *Wq = (const float*)d_in[4],  *bq = (const float*)d_in[5];
    const float *Wk = (const float*)d_in[6],  *bk = (const float*)d_in[7];
    const float *Wv = (const float*)d_in[8],  *bv = (const float*)d_in[9];
    const float *Wih = (const float*)d_in[10], *bih = (const float*)d_in[11];
    const float *Whh = (const float*)d_in[12], *bhh = (const float*)d_in[13];
    const float *Wo = (const float*)d_in[14], *bo = (const float*)d_in[15];
    const float *W1 = (const float*)d_in[16], *b1 = (const float*)d_in[17];
    const float *W2 = (const float*)d_in[18], *b2 = (const float*)d_in[19];
    const float *n1g = (const float*)d_in[20], *n1b = (const float*)d_in[21];
    const float *n3g = (const float*)d_in[22], *n3b = (const float*)d_in[23];
    float* out = (float*)d_out;

    // Workspace carve-up (256B aligned).
    size_t off = 0;
    auto carve = [&](size_t nfloats) {
        float* p = (float*)((char*)d_ws + off);
        off += ((nfloats * 4 + 255) / 256) * 256;
        return p;
    };
    float* xs    = carve((size_t)NSRC * DIM);
    float* Ks    = carve((size_t)NSRC * DIM);
    float* Vs    = carve((size_t)NSRC * DIM);
    float* maskS = carve((size_t)NSRC * NH);
    float* amax  = carve((size_t)NTGT * NH);
    float* denom = carve((size_t)NTGT * NH);
    float* alpha = carve((size_t)NEDGE * NH);
    float* B1    = carve((size_t)NTGT * DIM);   // Qt, later u
    float* B2    = carve((size_t)NTGT * DIM);   // Whh pre-gate, later LN(xt)
    float* agg   = carve((size_t)NTGT * DIM);
    float* Wq_p  = carve((size_t)DIM * DIM);
    float* Wk_p  = carve((size_t)DIM * DIM);
    float* Wv_p  = carve((size_t)DIM * DIM);
    float* Wih_p = carve((size_t)DIM * DIM);
    float* Whh_p = carve((size_t)DIM * DIM);
    float* Wo_p  = carve((size_t)DIM * DIM);
    float* W1_p  = carve((size_t)DIM * 4 * DIM);
    float* W2_p  = carve((size_t)4 * DIM * DIM);

    // Weight repack (cheap; weights reused by thousands of blocks).
    const int pb = 256;
    int pg_sq = (DIM * DIM / 2 + pb - 1) / pb;
    int pg_w1 = (DIM * 4 * DIM / 2 + pb - 1) / pb;
    pack_w<<<pg_sq, pb, 0, stream>>>(Wq,  Wq_p,  DIM, DIM);
    pack_w<<<pg_sq, pb, 0, stream>>>(Wk,  Wk_p,  DIM, DIM);
    pack_w<<<pg_sq, pb, 0, stream>>>(Wv,  Wv_p,  DIM, DIM);
    pack_w<<<pg_sq, pb, 0, stream>>>(Wih, Wih_p, DIM, DIM);
    pack_w<<<pg_sq, pb, 0, stream>>>(Whh, Whh_p, DIM, DIM);
    pack_w<<<pg_sq, pb, 0, stream>>>(Wo,  Wo_p,  DIM, DIM);
    pack_w<<<pg_w1, pb, 0, stream>>>(W1,  W1_p,  DIM, 4 * DIM);
    pack_w<<<pg_w1, pb, 0, stream>>>(W2,  W2_p,  4 * DIM, DIM);

    const long nAgg = (long)NTGT * DIM, nTH = (long)NTGT * NH;
    fill_f32<<<(int)((nAgg + 255) / 256), 256, 0, stream>>>(agg, 0.0f, nAgg);
    fill_f32<<<(int)((nTH + 255) / 256), 256, 0, stream>>>(denom, 0.0f, nTH);
    fill_f32<<<(int)((nTH + 255) / 256), 256, 0, stream>>>(amax, -__builtin_inff(), nTH);

    // Source side: LN, K/V projections, value mask (tiny).
    ln_rows<<<(NSRC + 7) / 8, 256, 0, stream>>>(x_src, n1g, n1b, xs, NSRC);
    int gS = (NSRC + 63) / 64;
    gemm_wmma_f32<0><<<gS, 256, 0, stream>>>(xs, Wk_p, bk, nullptr, nullptr, Ks, NSRC, DIM, DIM);
    gemm_wmma_f32<0><<<gS, 256, 0, stream>>>(xs, Wv_p, bv, nullptr, nullptr, Vs, NSRC, DIM, DIM);
    make_mask<<<(NSRC * NH + 255) / 256, 256, 0, stream>>>(Vs, maskS);

    // Target-side projections (the big GEMMs).
    int gT = (NTGT + 63) / 64;
    gemm_wmma_f32<0><<<gT, 256, 0, stream>>>(x_tgt, Wq_p, bq, nullptr, nullptr, B1, NTGT, DIM, DIM);
    gemm_wmma_f32<0><<<gT, 256, 0, stream>>>(x_tgt, Whh_p, bhh, nullptr, nullptr, B2, NTGT, DIM, DIM);

    // Edge passes: score+max, exp+sum, weighted scatter.
    edge_alpha<<<(NEDGE + 7) / 8, 256, 0, stream>>>(B1, Ks, esrc, edst, alpha, amax);
    edge_exp<<<(int)(((long)NEDGE * NH + 255) / 256), 256, 0, stream>>>(edst, alpha, amax, denom);
    edge_agg<<<(NEDGE + 7) / 8, 256, 0, stream>>>(esrc, edst, alpha, denom, maskS, Vs, agg);

    // Gate: u = agg * sigmoid(agg@Wih + bih + pre)  -> B1 (Qt dead).
    gemm_wmma_f32<3><<<gT, 256, 0, stream>>>(agg, Wih_p, bih, B2, agg, B1, NTGT, DIM, DIM);
    // xt = x_target + u@Wo + bo -> d_out.
    gemm_wmma_f32<1><<<gT, 256, 0, stream>>>(B1, Wo_p, bo, x_tgt, nullptr, out, NTGT, DIM, DIM);
    // h = LN(xt) -> B2; y = xt + MLP(h) -> d_out (fused, hidden in LDS).
    ln_rows<<<NTGT / 8, 256, 0, stream>>>(out, n3g, n3b, B2, NTGT);
    mlp_fused<<<(NTGT + 31) / 32, 256, 0, stream>>>(B2, W1_p, b1, W2_p, b2, out, NTGT);
}